// MetapathGATConv_13932873909204
// MI455X (gfx1250) — compile-verified
//
#include <hip/hip_runtime.h>

// ---------------- problem constants ----------------
#define N_NODES 16384
#define RREL 8
#define EMBED 256          // HEADS*OUT = 4*64 = 256 as well
#define NB 4               // nodes per workgroup
#define MROWS (NB * RREL)  // 32 rows of [*, 256] activations per block
#define LDW 260            // padded LDS row stride (floats): 1040B rows, 16B aligned, bank-spread
#define NEG_SLOPE 0.2f

typedef __attribute__((ext_vector_type(2))) float v2f;
typedef __attribute__((ext_vector_type(8))) float v8f;

static __device__ __forceinline__ v8f wmma_f32(v2f a, v2f b, v8f c) {
  // V_WMMA_F32_16X16X4_F32 : D = A(16x4) * B(4x16) + C, exact fp32
  return __builtin_amdgcn_wmma_f32_16x16x4_f32(
      /*neg_a=*/false, a, /*neg_b=*/false, b,
      /*c_mod=*/(short)0, c, /*reuse_a=*/false, /*reuse_b=*/false);
}

__global__ __launch_bounds__(128)
void metapath_gat_fused(const float* __restrict__ emb,
                        const float* __restrict__ Wl0, const float* __restrict__ bl0,
                        const float* __restrict__ Wr0, const float* __restrict__ br0,
                        const float* __restrict__ att0, const float* __restrict__ bias0,
                        const float* __restrict__ Wl1, const float* __restrict__ bl1,
                        const float* __restrict__ Wr1, const float* __restrict__ br1,
                        const float* __restrict__ att1, const float* __restrict__ bias1,
                        float* __restrict__ outEmb, float* __restrict__ outBeta) {
  __shared__ float sA[MROWS * LDW];   // h (layer input), later h1
  __shared__ float sXL[MROWS * LDW];  // xl, later xl1
  __shared__ float sXR[MROWS * LDW];  // xr, later xr1(self) in rows 0..15
  __shared__ float sAtt0[EMBED];
  __shared__ float sAtt1[EMBED];

  const int tid  = threadIdx.x;
  const int wave = tid >> 5;     // 0..3
  const int lane = tid & 31;
  const int l16  = lane & 15;    // M (A) / N (B,C,D) index within tile
  const int half = lane >> 4;    // selects K pair (A/B) or M+8 (C/D)
  const int nb   = blockIdx.x;
  const long rowBase = (long)nb * MROWS;  // row into the [N*R, 256] activation matrix

  // ---------------- Phase 0: load + ReLU input block into LDS ----------------
  for (int idx = tid; idx < (MROWS * EMBED) / 4; idx += 128) {
    const int elem = idx * 4;
    const int row  = elem >> 8;    // /256
    const int col  = elem & 255;
    const float4 v = *(const float4*)(emb + (rowBase + row) * (long)EMBED + col);
    float* dst = &sA[row * LDW + col];
    dst[0] = fmaxf(v.x, 0.f); dst[1] = fmaxf(v.y, 0.f);
    dst[2] = fmaxf(v.z, 0.f); dst[3] = fmaxf(v.w, 0.f);
  }
  for (int c = tid; c < EMBED; c += 128) { sAtt0[c] = att0[c]; sAtt1[c] = att1[c]; }
  __syncthreads();

  // ---------------- Phase 1: layer-0 GEMMs (fused K-loop) --------------------
  // xl = h@Wl0+bl0 and xr = h@Wr0+br0 share A fragments: 4 WMMAs per K-step.
  for (int t = 0; t < 4; ++t) {
    const int nt  = wave + t * 4;       // column tile 0..15
    const int col = nt * 16 + l16;
    v8f axl0 = {}, axl1 = {}, axr0 = {}, axr1 = {};
    for (int k0 = 0; k0 < EMBED; k0 += 4) {
      const int kk = k0 + half * 2;
      v2f bl, br;
      bl.x = Wl0[(size_t)kk * EMBED + col];
      bl.y = Wl0[(size_t)(kk + 1) * EMBED + col];
      br.x = Wr0[(size_t)kk * EMBED + col];
      br.y = Wr0[(size_t)(kk + 1) * EMBED + col];
      const v2f a0 = *(const v2f*)&sA[l16 * LDW + kk];
      const v2f a1 = *(const v2f*)&sA[(16 + l16) * LDW + kk];
      axl0 = wmma_f32(a0, bl, axl0);
      axl1 = wmma_f32(a1, bl, axl1);
      axr0 = wmma_f32(a0, br, axr0);
      axr1 = wmma_f32(a1, br, axr1);
    }
    const float bvl = bl0[col];
    const float bvr = br0[col];
#pragma unroll
    for (int v = 0; v < 8; ++v) {
      sXL[(v + half * 8) * LDW + col]      = axl0[v] + bvl;
      sXL[(16 + v + half * 8) * LDW + col] = axl1[v] + bvl;
      sXR[(v + half * 8) * LDW + col]      = axr0[v] + bvr;
      sXR[(16 + v + half * 8) * LDW + col] = axr1[v] + bvr;
    }
  }
  __syncthreads();

  // ---------------- Phase 2: layer-0 attention (wave w == node w) -------------
  {
    const int node = wave;              // 0..3
    const int j  = lane >> 2;           // source relation 0..7
    const int hh = lane & 3;            // head 0..3
    for (int i = 0; i < RREL; ++i) {    // destination relation
      const float* xr_row = &sXR[(node * RREL + i) * LDW + hh * 64];
      const float* xl_row = &sXL[(node * RREL + j) * LDW + hh * 64];
      const float* at     = &sAtt0[hh * 64];
      float logit = 0.f;
      for (int c = 0; c < 64; ++c) {
        const float s  = xr_row[c] + xl_row[c];
        const float lr = s > 0.f ? s : NEG_SLOPE * s;
        logit = fmaf(at[c], lr, logit);
      }
      // softmax over j: lanes differing in bits 2..4
      float m = logit;
      m = fmaxf(m, __shfl_xor(m, 4));
      m = fmaxf(m, __shfl_xor(m, 8));
      m = fmaxf(m, __shfl_xor(m, 16));
      const float e = __expf(logit - m);
      float ssum = e;
      ssum += __shfl_xor(ssum, 4);
      ssum += __shfl_xor(ssum, 8);
      ssum += __shfl_xor(ssum, 16);
      const float alpha = e / ssum;
      // h1[i] = relu(sum_j alpha * xl[j] + bias0)
      for (int u = 0; u < 8; ++u) {
        const int colc = lane + 32 * u;
        const int h2   = colc >> 6;
        float acc = 0.f;
#pragma unroll
        for (int jj = 0; jj < 8; ++jj) {
          const float a = __shfl(alpha, (jj << 2) | h2);
          acc = fmaf(a, sXL[(node * RREL + jj) * LDW + colc], acc);
        }
        acc += bias0[colc];
        sA[(node * RREL + i) * LDW + colc] = fmaxf(acc, 0.f);
      }
    }
  }
  __syncthreads();

  // ---------------- Phase 3: layer-1 GEMMs (fused K-loop) ---------------------
  // xl1 = h1@Wl1+bl1 (two 16-row tiles) and xr1 = h1[self]@Wr1+br1 (one padded
  // tile; rows p map to h1[node p&3][r=7], only rows 0..NB-1 meaningful).
  for (int t = 0; t < 4; ++t) {
    const int nt  = wave + t * 4;
    const int col = nt * 16 + l16;
    const int srow = (l16 & (NB - 1)) * RREL + (RREL - 1);
    v8f al0 = {}, al1 = {}, ar = {};
    for (int k0 = 0; k0 < EMBED; k0 += 4) {
      const int kk = k0 + half * 2;
      v2f bl, br;
      bl.x = Wl1[(size_t)kk * EMBED + col];
      bl.y = Wl1[(size_t)(kk + 1) * EMBED + col];
      br.x = Wr1[(size_t)kk * EMBED + col];
      br.y = Wr1[(size_t)(kk + 1) * EMBED + col];
      const v2f a0 = *(const v2f*)&sA[l16 * LDW + kk];
      const v2f a1 = *(const v2f*)&sA[(16 + l16) * LDW + kk];
      const v2f as = *(const v2f*)&sA[srow * LDW + kk];
      al0 = wmma_f32(a0, bl, al0);
      al1 = wmma_f32(a1, bl, al1);
      ar  = wmma_f32(as, br, ar);
    }
    const float bvl = bl1[col];
    const float bvr = br1[col];
#pragma unroll
    for (int v = 0; v < 8; ++v) {
      sXL[(v + half * 8) * LDW + col]      = al0[v] + bvl;
      sXL[(16 + v + half * 8) * LDW + col] = al1[v] + bvl;
      sXR[(v + half * 8) * LDW + col]      = ar[v] + bvr;  // rows 0..15; >=NB unused
    }
  }
  __syncthreads();

  // ---------------- Phase 4: layer-1 attention + outputs ----------------------
  {
    const int node = wave;                         // 0..3
    const long gn  = (long)nb * NB + node;         // global node id
    const int r  = lane >> 2;                      // relation 0..7
    const int hh = lane & 3;                       // head 0..3
    const float* xr_row = &sXR[node * LDW + hh * 64];              // xr1[self]
    const float* xl_row = &sXL[(node * RREL + r) * LDW + hh * 64]; // xl1[r]
    const float* at     = &sAtt1[hh * 64];
    float logit = 0.f;
    for (int c = 0; c < 64; ++c) {
      const float s  = xr_row[c] + xl_row[c];
      const float lr = s > 0.f ? s : NEG_SLOPE * s;
      logit = fmaf(at[c], lr, logit);
    }
    float m = logit;
    m = fmaxf(m, __shfl_xor(m, 4));
    m = fmaxf(m, __shfl_xor(m, 8));
    m = fmaxf(m, __shfl_xor(m, 16));
    const float e = __expf(logit - m);
    float ssum = e;
    ssum += __shfl_xor(ssum, 4);
    ssum += __shfl_xor(ssum, 8);
    ssum += __shfl_xor(ssum, 16);
    const float alpha = e / ssum;                  // betas, lane = r*4+h
    outBeta[gn * (RREL * 4) + lane] = alpha;

    for (int u = 0; u < 8; ++u) {
      const int colc = lane + 32 * u;
      const int h2   = colc >> 6;
      float acc = 0.f;
#pragma unroll
      for (int rr = 0; rr < 8; ++rr) {
        const float a = __shfl(alpha, (rr << 2) | h2);
        acc = fmaf(a, sXL[(node * RREL + rr) * LDW + colc], acc);
      }
      acc += bias1[colc];
      outEmb[gn * EMBED + colc] = fmaxf(acc, 0.f);
    }
  }
}

extern "C" void kernel_launch(void* const* d_in, const int* in_sizes, int n_in,
                              void* d_out, int out_size, void* d_ws, size_t ws_size,
                              hipStream_t stream) {
  (void)in_sizes; (void)n_in; (void)out_size; (void)d_ws; (void)ws_size;
  const float* emb   = (const float*)d_in[0];
  const float* Wl0   = (const float*)d_in[1];
  const float* bl0   = (const float*)d_in[2];
  const float* Wr0   = (const float*)d_in[3];
  const float* br0   = (const float*)d_in[4];
  const float* att0  = (const float*)d_in[5];
  const float* bias0 = (const float*)d_in[6];
  const float* Wl1   = (const float*)d_in[7];
  const float* bl1   = (const float*)d_in[8];
  const float* Wr1   = (const float*)d_in[9];
  const float* br1   = (const float*)d_in[10];
  const float* att1  = (const float*)d_in[11];
  const float* bias1 = (const float*)d_in[12];

  float* outEmb  = (float*)d_out;                              // [N, 256]
  float* outBeta = (float*)d_out + (size_t)N_NODES * EMBED;    // [N, 8, 4]

  dim3 grid(N_NODES / NB);   // 4096 blocks
  dim3 block(128);           // 4 waves (wave32)
  metapath_gat_fused<<<grid, block, 0, stream>>>(
      emb, Wl0, bl0, Wr0, br0, att0, bias0,
      Wl1, bl1, Wr1, br1, att1, bias1, outEmb, outBeta);
}